// CTCdecoder_74766790689111
// MI455X (gfx1250) — compile-verified
//
#include <hip/hip_runtime.h>
#include <math.h>

typedef __attribute__((ext_vector_type(2))) float v2f;
typedef __attribute__((ext_vector_type(8))) float v8f;

#define LOGZERO (-4290774016.0f) /* -(65504^2) */

#define B_  32
#define T_  512
#define D_  512
#define V_  4096
#define CB_ 30
#define LY_ 4
#define NSPLIT 8                 /* V-dimension splits per row */
#define NCOLS  (V_ / NSPLIT)     /* 512 columns per split */
#define NROWS (B_ * T_)          /* 16384 */

__device__ __forceinline__ float laddexp(float a, float b) {
    float mx = fmaxf(a, b);
    float mn = fminf(a, b);
    return mx + log1pf(expf(mn - mx));
}

// ---------------------------------------------------------------------------
// Kernel 1: fused GEMM (x @ W^T + b) -> per-(row, split) partial Sum exp(u).
// 8 waves/block = 4 M-tiles x 2 V-splits (waves sharing a split reuse W lines
// through WGP$).  v_wmma_f32_16x16x4_f32 over K=512, two 16-col N-tiles per
// sweep sharing one A fragment.  Fragment layouts per ISA 7.12.2:
//   A: lane(hi,lo) -> x[row=lo][k+2hi .. k+2hi+1]      (one b64 load)
//   B: lane(hi,lo) -> W[col=lo][k+2hi .. k+2hi+1]
//   C: VGPR j, lane(hi,lo) -> (row j+8*hi, col lo)
// Fixed-pivot softmax: s += exp(u); lse = log(s) later (logits are O(1)).
// ---------------------------------------------------------------------------
__global__ __launch_bounds__(256) void k_gemm_partial(
    const float* __restrict__ x, const float* __restrict__ W,
    const float* __restrict__ bias, float* __restrict__ part)
{
    const int tid  = threadIdx.x;
    const int lane = tid & 31;
    const int lo   = lane & 15;
    const int hi   = lane >> 4;
    const int wave = tid >> 5;
    const int mtile = (blockIdx.x >> 2) * 4 + (wave & 3);  // 0..1023
    const int split = (blockIdx.x & 3) * 2 + (wave >> 2);  // 0..7
    const int rowBase = mtile * 16;
    const int ntBase  = split * NCOLS;

    const float* xrow = x + (size_t)(rowBase + lo) * D_ + 2 * hi;

    float s[8];
#pragma unroll
    for (int j = 0; j < 8; ++j) s[j] = 0.0f;

    for (int nt = ntBase; nt < ntBase + NCOLS; nt += 32) {
        const float* w0 = W + (size_t)(nt + lo) * D_ + 2 * hi;
        const float* w1 = w0 + (size_t)16 * D_;
        // prefetch next N-group's W rows (speculative; dropped if OOB)
        __builtin_prefetch(w0 + (size_t)32 * D_, 0, 0);
        __builtin_prefetch(w1 + (size_t)32 * D_, 0, 0);
        v8f acc0 = {};
        v8f acc1 = {};
#pragma unroll 8
        for (int k = 0; k < D_; k += 4) {
            v2f a  = *reinterpret_cast<const v2f*>(xrow + k);
            v2f b0 = *reinterpret_cast<const v2f*>(w0 + k);
            v2f b1 = *reinterpret_cast<const v2f*>(w1 + k);
            acc0 = __builtin_amdgcn_wmma_f32_16x16x4_f32(
                false, a, false, b0, (short)0, acc0, false, false);
            acc1 = __builtin_amdgcn_wmma_f32_16x16x4_f32(
                false, a, false, b1, (short)0, acc1, false, false);
        }
        float bias0 = bias[nt + lo];
        float bias1 = bias[nt + 16 + lo];
#pragma unroll
        for (int j = 0; j < 8; ++j) {
            s[j] += expf(acc0[j] + bias0);
            s[j] += expf(acc1[j] + bias1);
        }
    }
    // sum partials across the 16 lanes holding different columns of each row
#pragma unroll
    for (int j = 0; j < 8; ++j) {
        float ss = s[j];
#pragma unroll
        for (int off = 1; off < 16; off <<= 1)
            ss += __shfl_xor(ss, off, 16);
        if (lo == 0) {
            int r = rowBase + j + 8 * hi;
            part[(size_t)r * NSPLIT + split] = ss;
        }
    }
}

// ---------------------------------------------------------------------------
// Kernel 2: recompute only the 31 needed log-probs per row (30 beam + blank),
// folding the 8 split-partials into lse = log(sum).
// ---------------------------------------------------------------------------
__global__ __launch_bounds__(256) void k_pick(
    const float* __restrict__ x, const float* __restrict__ W,
    const float* __restrict__ bias, const float* __restrict__ part,
    const int* __restrict__ beam, const int* __restrict__ pblank,
    float* __restrict__ Xn, float* __restrict__ blank_lp)
{
    const int tid = threadIdx.x;
    const int col = tid & 31;            // 0..29 beam, 30 blank, 31 idle
    const int rl  = tid >> 5;
    const int r   = blockIdx.x * 8 + rl; // global (b,t) row
    if (col > CB_) return;
    const int b = r / T_;
    const int vcol = (col < CB_) ? beam[b * CB_ + col] : *pblank;

    float ssum = 0.0f;
#pragma unroll
    for (int i = 0; i < NSPLIT; ++i) ssum += part[(size_t)r * NSPLIT + i];
    const float lse = logf(ssum);

    const float4* xr = reinterpret_cast<const float4*>(x + (size_t)r * D_);
    const float4* wc = reinterpret_cast<const float4*>(W + (size_t)vcol * D_);
    float acc = 0.0f;
#pragma unroll 4
    for (int k = 0; k < D_ / 4; ++k) {
        float4 a = xr[k];
        float4 w = wc[k];
        acc += a.x * w.x + a.y * w.y + a.z * w.z + a.w * w.w;
    }
    float val = acc + bias[vcol] - lse;
    if (col < CB_) Xn[(size_t)r * CB_ + col] = val;
    else           blank_lp[r] = val;
}

// ---------------------------------------------------------------------------
// Kernel 3: per-batch inclusive cumsum of blank log-probs (lastP1) + eos term.
// ---------------------------------------------------------------------------
__global__ __launch_bounds__(512) void k_cumsum(
    const float* __restrict__ blank_lp, const int* __restrict__ xl,
    float* __restrict__ last1, float* __restrict__ eosval)
{
    __shared__ float buf[T_];
    const int b = blockIdx.x;
    const int t = threadIdx.x;
    buf[t] = blank_lp[b * T_ + t];
    __syncthreads();
    for (int off = 1; off < T_; off <<= 1) {
        float v = buf[t];
        if (t >= off) v += buf[t - off];
        __syncthreads();
        buf[t] = v;
        __syncthreads();
    }
    last1[b * T_ + t] = buf[t];
    if (t == 0) {
        int X = xl[b];
        eosval[b] = (X >= 1 && X <= T_) ? buf[X - 1] : 0.0f;
    }
}

// ---------------------------------------------------------------------------
// Kernel 4: the CTC recurrence — 960 independent (b,c) lanes, 508 steps,
// with online logsumexp over t (invalid t contributes LOGZERO, as reference).
// ---------------------------------------------------------------------------
__global__ __launch_bounds__(128) void k_scan(
    const float* __restrict__ last1, const float* __restrict__ Xn,
    const float* __restrict__ blank_lp, const int* __restrict__ xl,
    const int* __restrict__ y, const int* __restrict__ beam,
    float* __restrict__ curP)
{
    const int gid = blockIdx.x * blockDim.x + threadIdx.x;
    if (gid >= B_ * CB_) return;
    const int b = gid / CB_;
    const int c = gid % CB_;
    const bool same = (y[b * LY_ + (LY_ - 1)] == beam[b * CB_ + c]);
    const int X = xl[b];
    const int start = (LY_ < T_ - 1) ? LY_ : (T_ - 1);
    const int loop_start = (start > 1) ? start : 1;

    float Pn = LOGZERO, Pb = LOGZERO;
    float M = LOGZERO, S = 0.0f;

    for (int t = loop_start; t < T_; ++t) {
        float l1   = last1[b * T_ + (t - 1)];
        float pref = same ? l1 : laddexp(LOGZERO, l1);
        float xn   = Xn[(size_t)(b * T_ + t) * CB_ + c];
        float xb   = blank_lp[b * T_ + t];
        float Pn_new = laddexp(Pn, pref) + xn;
        float Pb_new = laddexp(Pn, Pb) + xb;
        Pn = Pn_new;
        Pb = Pb_new;
        float cur = (X > t) ? laddexp(Pn, Pb) : LOGZERO;
        float nm = fmaxf(M, cur);
        S = S * expf(M - nm) + expf(cur - nm);
        M = nm;
    }
    curP[gid] = M + logf(S);
}

// ---------------------------------------------------------------------------
// Kernel 5a/5b: assemble finalP with the reference's override order.
// ---------------------------------------------------------------------------
__global__ void k_fill(float* __restrict__ out) {
    int i = blockIdx.x * blockDim.x + threadIdx.x;
    if (i < B_ * V_) out[i] = LOGZERO;
}

__global__ void k_scatter(const float* __restrict__ curP,
                          const float* __restrict__ eosval,
                          const int* __restrict__ beam,
                          const int* __restrict__ pblank,
                          const int* __restrict__ peos,
                          float* __restrict__ out)
{
    const int b = blockIdx.x;
    const int c = threadIdx.x; // 32 threads
    if (c < CB_)
        out[(size_t)b * V_ + beam[b * CB_ + c]] = curP[b * CB_ + c];
    __syncthreads();
    if (c == 0) {
        out[(size_t)b * V_ + *peos]   = eosval[b]; // eos after beam (override)
        out[(size_t)b * V_ + *pblank] = LOGZERO;   // blank last
    }
}

extern "C" void kernel_launch(void* const* d_in, const int* in_sizes, int n_in,
                              void* d_out, int out_size, void* d_ws, size_t ws_size,
                              hipStream_t stream)
{
    (void)in_sizes; (void)n_in; (void)out_size; (void)ws_size;
    const float* x      = (const float*)d_in[0];
    const float* W      = (const float*)d_in[1];
    const float* bias   = (const float*)d_in[2];
    const int*   xl     = (const int*)d_in[3];
    const int*   y      = (const int*)d_in[4];
    const int*   beam   = (const int*)d_in[5];
    const int*   pblank = (const int*)d_in[6];
    const int*   peos   = (const int*)d_in[7];
    float* out = (float*)d_out;

    float* part     = (float*)d_ws;                    // NROWS * NSPLIT
    float* blank_lp = part + (size_t)NROWS * NSPLIT;   // NROWS
    float* last1    = blank_lp + NROWS;                // NROWS
    float* Xn       = last1 + NROWS;                   // NROWS * CB
    float* eosv     = Xn + (size_t)NROWS * CB_;        // B
    float* curP     = eosv + B_;                       // B * CB

    // 1024 blocks x 8 waves: (M-tiles/4) x (splits/2) arrangement
    k_gemm_partial<<<(NROWS / 16 / 4) * (NSPLIT / 2), 256, 0, stream>>>(x, W, bias, part);
    k_pick<<<NROWS / 8, 256, 0, stream>>>(x, W, bias, part, beam, pblank, Xn, blank_lp);
    k_cumsum<<<B_, T_, 0, stream>>>(blank_lp, xl, last1, eosv);
    k_scan<<<(B_ * CB_ + 127) / 128, 128, 0, stream>>>(last1, Xn, blank_lp, xl, y, beam, curP);
    k_fill<<<(B_ * V_ + 255) / 256, 256, 0, stream>>>(out);
    k_scatter<<<B_, 32, 0, stream>>>(curP, eosv, beam, pblank, peos, out);
}